// CustomReshapeLayer_17111149707839
// MI455X (gfx1250) — compile-verified
//
#include <hip/hip_runtime.h>
#include <hip/hip_bf16.h>

// Bandwidth-bound triu scatter: out[b, r, c] = (c >= r) ? in[b, off(r) + c - r] : 0
// off(r) = r*1024 - r*(r-1)/2.
//
// Mapping: one wave32 per matrix row (wave id = b*1024 + r). A row is 256
// float4 chunks = 8 wave-widths; lane L handles chunks {32k + L}. Row/offset
// math is computed once per wave and amortized over 128 B of stores. Every
// load/store instruction covers a contiguous 512 B span; the diagonal
// classification only diverges in the iteration covering column r. Zero chunks issue no VMEM loads.
// Streams (269 MB in, 537 MB out) exceed the 192 MB L2, so all traffic is NT.
// Roofline: ~806 MB @ 23.3 TB/s ~= 35 us. 0 FLOPs: no matrix op applies.

typedef __attribute__((ext_vector_type(4))) float v4f;

#define MATSIZE   1024u
#define TRIU_LEN  524800u   // 1024*1025/2
#define NBATCH    128u

__global__ __launch_bounds__(256) void triu_scatter_kernel(
    const float* __restrict__ in, float* __restrict__ out)
{
    const unsigned gtid = blockIdx.x * 256u + threadIdx.x;
    const unsigned wid  = gtid >> 5;          // global wave id == (b*1024 + r)
    const unsigned lane = threadIdx.x & 31u;

    const unsigned b = wid >> 10;             // batch
    const unsigned r = wid & 1023u;           // row (wave-uniform)

    const float* __restrict__ inb = in + (size_t)b * TRIU_LEN;

    // off(r) = r*1024 - r*(r-1)/2 : start of row r inside triu storage.
    const unsigned off    = r * MATSIZE - ((r * (r - 1u)) >> 1);
    const unsigned inbase = off - r;          // input index of column c is inbase + c

    // Output row start, in float4 units: (b*1024 + r) * 256.
    v4f* __restrict__ outrow = (v4f*)out + ((size_t)wid << 8);

#pragma unroll
    for (unsigned k = 0; k < 8u; ++k) {
        const unsigned chunk = k * 32u + lane;   // float4 index within the row
        const unsigned c0    = chunk << 2;       // first column of this chunk

        v4f v;
        if (c0 >= r) {
            // Entire chunk in the upper triangle: unconditional streaming loads.
            const float* __restrict__ s = inb + inbase + c0;
            v.x = __builtin_nontemporal_load(s + 0);
            v.y = __builtin_nontemporal_load(s + 1);
            v.z = __builtin_nontemporal_load(s + 2);
            v.w = __builtin_nontemporal_load(s + 3);
        } else if (c0 + 3u < r) {
            // Entirely below the diagonal: zero fill, no loads.
            v.x = 0.0f; v.y = 0.0f; v.z = 0.0f; v.w = 0.0f;
        } else {
            // Diagonal straddle: one chunk per row.
#pragma unroll
            for (int i = 0; i < 4; ++i) {
                const unsigned c = c0 + (unsigned)i;
                v[i] = (c >= r) ? __builtin_nontemporal_load(inb + inbase + c)
                                : 0.0f;
            }
        }

        // Aligned 16B streaming store (global_store_b128, TH=NT).
        __builtin_nontemporal_store(v, outrow + chunk);
    }
}

extern "C" void kernel_launch(void* const* d_in, const int* in_sizes, int n_in,
                              void* d_out, int out_size, void* d_ws, size_t ws_size,
                              hipStream_t stream) {
    (void)in_sizes; (void)n_in; (void)out_size; (void)d_ws; (void)ws_size;

    const float* in = (const float*)d_in[0];      // [128, 524800] f32
    float*      out = (float*)d_out;              // [128, 1024, 1024] f32

    // One wave32 per row: 128 * 1024 waves, 8 waves per 256-thread block.
    const unsigned total_waves = NBATCH * MATSIZE;       // 131,072
    const unsigned blocks      = total_waves / 8u;       // 16,384

    triu_scatter_kernel<<<blocks, 256, 0, stream>>>(in, out);
}